// GCN_46712064311554
// MI455X (gfx1250) — compile-verified
//
#include <hip/hip_runtime.h>
#include <stdint.h>

#define IN_F  128
#define HID_F 128
#define OUT_F 64

typedef __attribute__((ext_vector_type(2))) float v2f;
typedef __attribute__((ext_vector_type(8))) float v8f;

// ---------------------------------------------------------------- zero fill
__global__ void zero_f32(float* __restrict__ p, long long n) {
  long long i = (long long)blockIdx.x * blockDim.x + threadIdx.x;
  long long stride = (long long)gridDim.x * blockDim.x;
  for (; i < n; i += stride) p[i] = 0.0f;
}

// ---------------------------------------------------------------- degree counts
__global__ void count_edges(const long long* __restrict__ dst,
                            float* __restrict__ cnt, int E) {
  int e = blockIdx.x * blockDim.x + threadIdx.x;
  if (e < E) atomicAdd(&cnt[(size_t)dst[e]], 1.0f);
}

// ------------------------------------------------ edge gather + atomic scatter
// One wave covers F floats of one (or two, for F=64) edges: lane chunk = float4.
__global__ void scatter_edges(const float* __restrict__ h, float* __restrict__ aggr,
                              const long long* __restrict__ src,
                              const long long* __restrict__ dst,
                              int E, int F) {
  const int chunks = F >> 2;  // float4 chunks per edge
  long long t = (long long)blockIdx.x * blockDim.x + threadIdx.x;
  if (t >= (long long)E * chunks) return;
  const int e = (int)(t / chunks);
  const int c = (int)(t % chunks);
  const size_t s = (size_t)src[e];
  const size_t d = (size_t)dst[e];
  const float4 val = *((const float4*)(h + s * (size_t)F) + c);
  float* ap = aggr + d * (size_t)F + (size_t)c * 4;
  atomicAdd(ap + 0, val.x);
  atomicAdd(ap + 1, val.y);
  atomicAdd(ap + 2, val.z);
  atomicAdd(ap + 3, val.w);
}

// ---------------------------------------------------------------- plain GEMM
// C[N x F] = A[N x K] @ B[K x F], all row-major.
// block = 32 * (F/16) threads (one wave per 16x16 col tile), grid.x = N/16.
__global__ void gemm_wmma_f32(const float* __restrict__ A, const float* __restrict__ B,
                              float* __restrict__ C, int K, int F) {
  const int lane    = threadIdx.x & 31;
  const int colTile = threadIdx.x >> 5;
  const int rowTile = blockIdx.x;
  const int half    = lane >> 4;   // 0: K pair {0,1}; 1: K pair {2,3}
  const int l16     = lane & 15;

  const float* Arow = A + (size_t)(rowTile * 16 + l16) * K;   // A: M = lane%16
  const float* Bcol = B + (size_t)(colTile * 16 + l16);       // B: N = lane%16

  v8f acc = {};
  for (int k0 = 0; k0 < K; k0 += 4) {
    v2f a = *(const v2f*)(Arow + k0 + half * 2);
    v2f b;
    b.x = Bcol[(size_t)(k0 + half * 2 + 0) * F];
    b.y = Bcol[(size_t)(k0 + half * 2 + 1) * F];
    acc = __builtin_amdgcn_wmma_f32_16x16x4_f32(false, a, false, b,
                                                (short)0, acc, false, false);
  }
  // C/D layout: element v -> row = rowTile*16 + v + 8*half, col = colTile*16 + l16
  float* Cp = C + (size_t)(rowTile * 16 + half * 8) * F + colTile * 16 + l16;
#pragma unroll
  for (int v = 0; v < 8; ++v) Cp[(size_t)v * F] = acc[v];
}

// ------------------------------------- skip GEMM fused with GraphConv epilogue
// Out = relu?( aggr/max(cnt,1) + A@B + bias )
__global__ void combine_wmma_f32(const float* __restrict__ A, const float* __restrict__ B,
                                 const float* __restrict__ bias,
                                 const float* __restrict__ aggr,
                                 const float* __restrict__ cnt,
                                 float* __restrict__ Out, int K, int F, int do_relu) {
  const int lane    = threadIdx.x & 31;
  const int colTile = threadIdx.x >> 5;
  const int rowTile = blockIdx.x;
  const int half    = lane >> 4;
  const int l16     = lane & 15;

  const float* Arow = A + (size_t)(rowTile * 16 + l16) * K;
  const float* Bcol = B + (size_t)(colTile * 16 + l16);

  v8f acc = {};
  for (int k0 = 0; k0 < K; k0 += 4) {
    v2f a = *(const v2f*)(Arow + k0 + half * 2);
    v2f b;
    b.x = Bcol[(size_t)(k0 + half * 2 + 0) * F];
    b.y = Bcol[(size_t)(k0 + half * 2 + 1) * F];
    acc = __builtin_amdgcn_wmma_f32_16x16x4_f32(false, a, false, b,
                                                (short)0, acc, false, false);
  }
  const int row0 = rowTile * 16 + half * 8;
  const int col  = colTile * 16 + l16;
  const float bv = bias[col];
#pragma unroll
  for (int v = 0; v < 8; ++v) {
    const int row   = row0 + v;
    const float inv = 1.0f / fmaxf(cnt[row], 1.0f);
    float d = acc[v] + aggr[(size_t)row * F + col] * inv + bv;
    if (do_relu) d = fmaxf(d, 0.0f);
    Out[(size_t)row * F + col] = d;
  }
}

// -----------------------------------------------------------------------------
extern "C" void kernel_launch(void* const* d_in, const int* in_sizes, int n_in,
                              void* d_out, int out_size, void* d_ws, size_t ws_size,
                              hipStream_t stream) {
  const float*      x   = (const float*)d_in[0];
  const long long*  ei  = (const long long*)d_in[1];   // int64 [2, E]
  const float*      W1  = (const float*)d_in[2];
  const float*      l1w = (const float*)d_in[3];
  const float*      l1b = (const float*)d_in[4];
  const float*      W2  = (const float*)d_in[5];
  const float*      l2w = (const float*)d_in[6];
  const float*      l2b = (const float*)d_in[7];
  float*            out = (float*)d_out;

  const int N = in_sizes[0] / IN_F;   // 50000 (divisible by 16)
  const int E = in_sizes[1] / 2;      // 800000
  const long long* src = ei;
  const long long* dst = ei + E;

  // workspace layout (floats): hbuf[N*128] | aggr[N*128] | cnt[N] | act1[N*128]
  float* ws   = (float*)d_ws;
  float* hbuf = ws;
  float* aggr = hbuf + (size_t)N * HID_F;
  float* cnt  = aggr + (size_t)N * HID_F;
  float* act1 = cnt + N;

  const int rowTiles = N / 16;

  // ---- layer 1 ----
  {  // zero aggr (N*128) + cnt (N): contiguous
    long long nz = (long long)N * HID_F + N;
    int blocks = (int)((nz + 255) / 256); if (blocks > 65535) blocks = 65535;
    zero_f32<<<blocks, 256, 0, stream>>>(aggr, nz);
  }
  gemm_wmma_f32<<<rowTiles, 32 * (HID_F / 16), 0, stream>>>(x, W1, hbuf, IN_F, HID_F);
  count_edges<<<(E + 255) / 256, 256, 0, stream>>>(dst, cnt, E);
  {
    long long tot = (long long)E * (HID_F / 4);
    scatter_edges<<<(int)((tot + 255) / 256), 256, 0, stream>>>(hbuf, aggr, src, dst, E, HID_F);
  }
  combine_wmma_f32<<<rowTiles, 32 * (HID_F / 16), 0, stream>>>(
      x, l1w, l1b, aggr, cnt, act1, IN_F, HID_F, /*relu=*/1);

  // ---- layer 2 ----
  {  // zero aggr (N*64)
    long long nz = (long long)N * OUT_F;
    int blocks = (int)((nz + 255) / 256); if (blocks > 65535) blocks = 65535;
    zero_f32<<<blocks, 256, 0, stream>>>(aggr, nz);
  }
  gemm_wmma_f32<<<rowTiles, 32 * (OUT_F / 16), 0, stream>>>(act1, W2, hbuf, HID_F, OUT_F);
  {
    long long tot = (long long)E * (OUT_F / 4);
    scatter_edges<<<(int)((tot + 255) / 256), 256, 0, stream>>>(hbuf, aggr, src, dst, E, OUT_F);
  }
  combine_wmma_f32<<<rowTiles, 32 * (OUT_F / 16), 0, stream>>>(
      act1, l2w, l2b, aggr, cnt, out, HID_F, OUT_F, /*relu=*/0);
}